// BERT_CRF_NER_7490422964891
// MI455X (gfx1250) — compile-verified
//
#include <hip/hip_runtime.h>

typedef __attribute__((ext_vector_type(2))) float v2f;
typedef __attribute__((ext_vector_type(8))) float v8f;

#define B_     128
#define T_     2048
#define L_     64
#define NEGV   (-10000.0f)
#define COLS   16     // batch columns handled per workgroup
#define FPITCH 68     // floats per featR row: 16B-aligned, bank-skewed

// ---------------------------------------------------------------------------
// Forward algorithm: sequential scan over t, one workgroup per 16 batch cols.
// Per step:  S(64x16) = exp(trans)(64x64) @ exp(alpha - colmax)(64x16)
// computed with chained V_WMMA_F32_16X16X4_F32 (4 waves x 16 K-chunks).
// Feat tiles stream in via async global->LDS (ASYNCcnt), double-buffered,
// with per-wave ownership so no barrier is needed on the feat buffer.
// ---------------------------------------------------------------------------
__global__ __launch_bounds__(128)
void crf_forward_kernel(const float* __restrict__ feats,
                        const float* __restrict__ trans,
                        float* __restrict__ fwd_out)
{
    __shared__ float Esh[L_][L_ + 1];       // exp(transitions), padded
    __shared__ float alpha[L_][COLS + 1];   // CRF state, [tag][batch-col]
    __shared__ float P[L_][COLS + 1];       // exp(alpha - colmax)
    __shared__ __align__(16) float featR[2][COLS][FPITCH]; // [buf][col][tag]

    const int tid   = threadIdx.x;
    const int b0    = blockIdx.x * COLS;
    const int lane  = tid & 31;
    const int wave  = tid >> 5;     // 0..3 -> 16-row tile of D
    const int rlo   = lane & 15;
    const int h     = lane >> 4;
    const int row16 = wave * 16;

    // Async feat pipeline: wave w owns tag rows [16w,16w+16) for all 16 cols.
    // Lane (rlo,h) copies (col=rlo, tags l0..l0+7) as two b128 chunks.
    const int l0 = row16 + 8 * h;
    const unsigned lds0 = (unsigned)(size_t)&featR[0][rlo][l0];
    const unsigned lds1 = (unsigned)(size_t)&featR[1][rlo][l0];
    const float* gptr = feats + (((size_t)(b0 + rlo) * T_ + 1) * L_) + l0;

    // Prefetch t=1 tile into buffer 1 (overlaps with E/alpha init below).
    asm volatile("global_load_async_to_lds_b128 %0, %1, off"
                 :: "v"(lds1), "v"(gptr) : "memory");
    asm volatile("global_load_async_to_lds_b128 %0, %1, off offset:16"
                 :: "v"(lds1), "v"(gptr) : "memory");
    gptr += L_;   // -> t=2

    // E = exp(transitions)  (time-invariant GEMM operand)
    for (int idx = tid; idx < L_ * L_; idx += 128) {
        int i = idx >> 6, j = idx & 63;
        Esh[i][j] = __expf(trans[idx]);
    }
    // alpha init: START tag = 0, rest NEG
    for (int idx = tid; idx < L_ * COLS; idx += 128) {
        int row = idx >> 4, col = idx & 15;
        alpha[row][col] = (row == 0) ? 0.0f : NEGV;
    }
    __syncthreads();

    for (int t = 1; t < T_; ++t) {
        // ---- issue async copy of the t+1 feat tile into the other buffer ----
        if (t + 1 < T_) {
            unsigned lo = ((t + 1) & 1) ? lds1 : lds0;
            asm volatile("global_load_async_to_lds_b128 %0, %1, off"
                         :: "v"(lo), "v"(gptr) : "memory");
            asm volatile("global_load_async_to_lds_b128 %0, %1, off offset:16"
                         :: "v"(lo), "v"(gptr) : "memory");
            gptr += L_;
        }

        // ---- column max: each lane scans 32 rows of its column (half h),
        //      stashing the 8 alpha values of the P-rows this lane owns ----
        float stash[8];
        float m = -3.0e38f;
        #pragma unroll
        for (int seg = 0; seg < 4; ++seg) {
            #pragma unroll
            for (int j = 0; j < 8; ++j) {
                float v = alpha[32 * h + seg * 8 + j][rlo];
                m = fmaxf(m, v);
                if (seg == wave) stash[j] = v;
            }
        }
        m = fmaxf(m, __shfl_xor(m, 16, 32));   // merge halves: colmax[rlo]

        // ---- P rows owned by this lane: exp(alpha - colmax) ----
        #pragma unroll
        for (int j = 0; j < 8; ++j)
            P[32 * h + wave * 8 + j][rlo] = __expf(stash[j] - m);
        __syncthreads();

        // ---- GEMM: D = E @ P, 16 chained f32 WMMAs over K ----
        // A 16x4: lane = M, VGPRv + 2h = K offset; B 4x16: lane = N;
        // D 16x16: VGPR r holds rows {r, r+8} (half h), col = lane&15.
        v8f acc = {};
        #pragma unroll
        for (int k = 0; k < 16; ++k) {
            const int kb = k * 4 + 2 * h;
            v2f a, b;
            a.x = Esh[row16 + rlo][kb];
            a.y = Esh[row16 + rlo][kb + 1];
            b.x = P[kb][rlo];
            b.y = P[kb + 1][rlo];
            acc = __builtin_amdgcn_wmma_f32_16x16x4_f32(
                      false, a, false, b, (short)0, acc, false, false);
        }

        // ---- wait for this step's feat tile (keep next step's 2 in flight) ----
        if (t + 1 < T_) asm volatile("s_wait_asynccnt 0x2" ::: "memory");
        else            asm volatile("s_wait_asynccnt 0x0" ::: "memory");

        // ---- new alpha = log(S) + colmax + feat_t (wave-owned feat rows) ----
        const int p = t & 1;
        #pragma unroll
        for (int r = 0; r < 8; ++r) {
            int rowM = row16 + r + 8 * h;
            float v  = __logf(fmaxf(acc[r], 1e-30f)) + m + featR[p][rlo][rowM];
            alpha[rowM][rlo] = v;
        }
        __syncthreads();
    }

    // ---- forward_score[b] = logsumexp over tags ----
    if (tid < COLS) {
        float m = alpha[0][tid];
        for (int r = 1; r < L_; ++r) m = fmaxf(m, alpha[r][tid]);
        float s = 0.0f;
        for (int r = 0; r < L_; ++r) s += __expf(alpha[r][tid] - m);
        fwd_out[b0 + tid] = m + __logf(s);
    }
}

// ---------------------------------------------------------------------------
// Gold-path score: one block per batch, gather + tree reduce.
// ---------------------------------------------------------------------------
__global__ __launch_bounds__(256)
void crf_gold_kernel(const float* __restrict__ feats,
                     const float* __restrict__ trans,
                     const int* __restrict__ labels,
                     float* __restrict__ gold_out)
{
    __shared__ float red[256];
    const int b   = blockIdx.x;
    const int tid = threadIdx.x;
    float acc = 0.0f;
    for (int t = 1 + tid; t < T_; t += 256) {
        int cur  = labels[(size_t)b * T_ + t];
        int prev = labels[(size_t)b * T_ + t - 1];
        acc += trans[cur * L_ + prev] + feats[((size_t)b * T_ + t) * L_ + cur];
    }
    red[tid] = acc;
    __syncthreads();
    for (int s = 128; s > 0; s >>= 1) {
        if (tid < s) red[tid] += red[tid + s];
        __syncthreads();
    }
    if (tid == 0) gold_out[b] = red[0];
}

// ---------------------------------------------------------------------------
// Final: mean over batch of (forward - gold) -> scalar
// ---------------------------------------------------------------------------
__global__ __launch_bounds__(128)
void crf_final_kernel(const float* __restrict__ fwd,
                      const float* __restrict__ gold,
                      float* __restrict__ out)
{
    __shared__ float red[128];
    const int tid = threadIdx.x;
    red[tid] = fwd[tid] - gold[tid];
    __syncthreads();
    for (int s = 64; s > 0; s >>= 1) {
        if (tid < s) red[tid] += red[tid + s];
        __syncthreads();
    }
    if (tid == 0) out[0] = red[0] * (1.0f / B_);
}

extern "C" void kernel_launch(void* const* d_in, const int* in_sizes, int n_in,
                              void* d_out, int out_size, void* d_ws, size_t ws_size,
                              hipStream_t stream) {
    (void)in_sizes; (void)n_in; (void)out_size; (void)ws_size;
    const float* feats  = (const float*)d_in[0];   // (128, 2048, 64) f32
    const float* trans  = (const float*)d_in[1];   // (64, 64) f32
    const int*   labels = (const int*)d_in[2];     // (128, 2048) int

    float* fwd  = (float*)d_ws;        // 128 floats
    float* gold = fwd + B_;            // 128 floats

    crf_forward_kernel<<<B_ / COLS, 128, 0, stream>>>(feats, trans, fwd);
    crf_gold_kernel<<<B_, 256, 0, stream>>>(feats, trans, labels, gold);
    crf_final_kernel<<<1, 128, 0, stream>>>(fwd, gold, (float*)d_out);
}